// Nonlinearity_49074296324432
// MI455X (gfx1250) — compile-verified
//
#include <hip/hip_runtime.h>

typedef __attribute__((ext_vector_type(2))) float v2f;
typedef __attribute__((ext_vector_type(8))) float v8f;

namespace cgt {

constexpr int LMAX   = 4;
constexpr int NCOMBO = 42;
constexpr int NPAIR  = 15;
constexpr int NB     = 64;
constexpr int NC     = 32;

constexpr int cmin(int a, int b) { return a < b ? a : b; }

struct Tables {
  // combo data (reference COMBOS order: l1 outer, l2 mid, L inner)
  int comboL[NCOMBO];
  int comboL1[NCOMBO];
  int comboL2[NCOMBO];
  int comboPair[NCOMBO];
  int comboColBase[NCOMBO];
  int cgOff[NCOMBO + 1];
  // pair data (l1 <= l2, 15 pairs)
  int pairL1[NPAIR];
  int pairL2[NPAIR];
  int pairK[NPAIR];    // (2l1+1)(2l2+1)
  int pairKp[NPAIR];   // padded to multiple of 4
  int pairN[NPAIR];    // sum over L of (2L+1)
  int pairNp[NPAIR];   // padded to multiple of 16
  int pairCgw[NPAIR];  // float offset of this pair's CG matrix in workspace
  int cgwTotal;
  // per (pair, col) output mapping; all offsets in complex64 elements
  int colBase[NPAIR][32];   // base offset (excl. batch), -1 = padding col
  int colBstr[NPAIR][32];   // batch stride
  int colIJstr[NPAIR][32];  // ij stride  (= 2L+1)
};

constexpr Tables makeTables() {
  Tables t{};
  int k = 0;
  t.cgOff[0] = 0;
  for (int l1 = 0; l1 <= LMAX; ++l1)
    for (int l2 = l1; l2 <= LMAX; ++l2)
      for (int L = l2 - l1; L <= cmin(LMAX, l1 + l2); ++L) {
        t.comboL[k] = L; t.comboL1[k] = l1; t.comboL2[k] = l2;
        t.cgOff[k + 1] = t.cgOff[k] + (2*L+1) * (2*l2+1) * (2*l1+1);
        ++k;
      }
  int nL[LMAX + 1] = {};
  int slot[NCOMBO] = {};
  for (int kk = 0; kk < NCOMBO; ++kk) slot[kk] = nL[t.comboL[kk]]++;
  int baseL[LMAX + 1] = {};
  int acc = 0;
  for (int L = 0; L <= LMAX; ++L) { baseL[L] = acc; acc += NB * nL[L] * NC * NC * (2*L+1); }
  int p = 0, cgw = 0;
  for (int l1 = 0; l1 <= LMAX; ++l1)
    for (int l2 = l1; l2 <= LMAX; ++l2) {
      const int d1 = 2*l1+1, d2 = 2*l2+1;
      const int K  = d1 * d2;
      const int Kp = (K + 3) & ~3;
      int N = 0;
      for (int L = l2 - l1; L <= cmin(LMAX, l1 + l2); ++L) N += 2*L+1;
      const int Np = (N + 15) & ~15;
      t.pairL1[p] = l1; t.pairL2[p] = l2;
      t.pairK[p] = K; t.pairKp[p] = Kp; t.pairN[p] = N; t.pairNp[p] = Np;
      t.pairCgw[p] = cgw; cgw += Kp * Np;
      for (int c = 0; c < 32; ++c) { t.colBase[p][c] = -1; t.colBstr[p][c] = 0; t.colIJstr[p][c] = 0; }
      int col = 0;
      for (int L = l2 - l1; L <= cmin(LMAX, l1 + l2); ++L) {
        int kg = -1;
        for (int kk = 0; kk < NCOMBO; ++kk)
          if (t.comboL[kk] == L && t.comboL1[kk] == l1 && t.comboL2[kk] == l2) kg = kk;
        t.comboPair[kg] = p;
        t.comboColBase[kg] = col;
        for (int M = 0; M < 2*L+1; ++M) {
          t.colBase[p][col]  = baseL[L] + slot[kg] * NC * NC * (2*L+1) + M;
          t.colBstr[p][col]  = nL[L] * NC * NC * (2*L+1);
          t.colIJstr[p][col] = 2*L+1;
          ++col;
        }
      }
      ++p;
    }
  t.cgwTotal = cgw;
  return t;
}

constexpr Tables HT = makeTables();

} // namespace cgt

__constant__ cgt::Tables dT = cgt::makeTables();

// ---------------------------------------------------------------------------
// Prep: zero the padded CG workspace, then scatter cg_flat into per-pair
// column-major [col][k] matrices (k = n*d1 + m, col enumerates (L,M)).
// Zero padding in k means padded-k WMMA lanes contribute exactly 0.
// ---------------------------------------------------------------------------
__global__ void cgw_zero_kernel(float* __restrict__ cgw, int total) {
  int i = blockIdx.x * blockDim.x + threadIdx.x;
  if (i < total) cgw[i] = 0.0f;
}

__global__ void cgw_prep_kernel(const float* __restrict__ cg_flat, float* __restrict__ cgw) {
  const int kg = blockIdx.x;                       // one block per combo
  const int L  = dT.comboL[kg];
  const int d1 = 2 * dT.comboL1[kg] + 1;
  const int d2 = 2 * dT.comboL2[kg] + 1;
  const int p  = dT.comboPair[kg];
  const int Kp = dT.pairKp[p];
  const int colBase = dT.comboColBase[kg];
  const int sz = (2*L+1) * d2 * d1;
  const float* src = cg_flat + dT.cgOff[kg];
  float* dst = cgw + dT.pairCgw[p];
  for (int e = threadIdx.x; e < sz; e += blockDim.x) {
    const int M = e / (d2 * d1);
    const int r = e - M * d2 * d1;
    const int n = r / d1;
    const int m = r - n * d1;
    dst[(colBase + M) * Kp + n * d1 + m] = src[e];
  }
}

// ---------------------------------------------------------------------------
// Main: one kernel specialization per (l1<=l2) pair. One block per batch b.
// 8 waves; each wave owns row-tiles rt = wave, wave+8, ... (16 ij-rows each).
// x1/x2 rows live in registers (compile-time d1/d2), CG B-operand in LDS.
// Fully unrolled K loop of v_wmma_f32_16x16x4_f32 (re & im GEMMs, 1-2 col
// tiles). Output written with non-temporal 8-byte stores (write-once stream).
// ---------------------------------------------------------------------------
template <int P, int L1, int L2, int NCOLS, int CGO>
__global__ __launch_bounds__(256, 1) void cg_pair_kernel(
    const float2* __restrict__ x0, const float2* __restrict__ x1,
    const float2* __restrict__ x2, const float2* __restrict__ x3,
    const float2* __restrict__ x4, const float* __restrict__ cgw,
    float2* __restrict__ out)
{
  constexpr int d1 = 2 * L1 + 1;
  constexpr int d2 = 2 * L2 + 1;
  constexpr int K  = d1 * d2;
  constexpr int Kp = (K + 3) & ~3;
  constexpr int Np = (NCOLS + 15) & ~15;
  constexpr int Nt = Np >> 4;          // 1 or 2 column tiles

  __shared__ float2 sx1[32 * d1];
  __shared__ float2 sx2[32 * d2];
  __shared__ float  scg[Np * Kp];      // [col][k], Kp-strided, zero-padded

  const int b = blockIdx.x;
  const float2* X1 = (L1 == 0) ? x0 : (L1 == 1) ? x1 : (L1 == 2) ? x2 : (L1 == 3) ? x3 : x4;
  const float2* X2 = (L2 == 0) ? x0 : (L2 == 1) ? x1 : (L2 == 2) ? x2 : (L2 == 3) ? x3 : x4;
  X1 += (size_t)b * 32 * d1;
  X2 += (size_t)b * 32 * d2;

  const int tid = threadIdx.x;
  for (int e = tid; e < 32 * d1; e += 256) sx1[e] = X1[e];
  for (int e = tid; e < 32 * d2; e += 256) sx2[e] = X2[e];
  for (int e = tid; e < Np * Kp; e += 256) scg[e] = cgw[CGO + e];
  __syncthreads();

  const int lane = tid & 31;
  const int wave = tid >> 5;
  const int hi   = lane >> 4;          // 0: provides k0/k1, 1: provides k2/k3
  const int rl   = lane & 15;

  // per-column output metadata, hoisted to registers
  int cb[Nt], bstr[Nt], ijstr[Nt];
#pragma unroll
  for (int ct = 0; ct < Nt; ++ct) {
    const int col = (ct << 4) | rl;
    cb[ct]    = dT.colBase[P][col];
    bstr[ct]  = dT.colBstr[P][col];
    ijstr[ct] = dT.colIJstr[P][col];
  }

  // B operand base: scg[col][ks + 2*hi], ct/ks offsets become immediates
  const float* cgcol = &scg[rl * Kp + (hi << 1)];

#pragma unroll 1
  for (int rt = wave; rt < 64; rt += 8) {
    // ij = rt*16 + local row; i = ij>>5 is constant per tile, j varies by lane
    const int i = rt >> 1;
    const int j = ((rt & 1) << 4) | rl;

    // register-resident complex rows (compile-time sizes)
    float2 r1[d1], r2[d2];
#pragma unroll
    for (int m = 0; m < d1; ++m) r1[m] = sx1[i * d1 + m];
#pragma unroll
    for (int n = 0; n < d2; ++n) r2[n] = sx2[j * d2 + n];

    v8f accRe[Nt], accIm[Nt];
    const v8f vz = {0.f,0.f,0.f,0.f,0.f,0.f,0.f,0.f};
#pragma unroll
    for (int ct = 0; ct < Nt; ++ct) { accRe[ct] = vz; accIm[ct] = vz; }

#pragma unroll
    for (int ks = 0; ks < Kp; ks += 4) {
      v2f aRe, aIm;
#pragma unroll
      for (int t = 0; t < 2; ++t) {
        const int klo = ks + t;                      // k for lanes 0-15
        const int khi = ks + 2 + t;                  // k for lanes 16-31
        // padded k -> (0,0): finite garbage, multiplied by zero CG column
        const int mlo = (klo < K) ? (klo % d1) : 0;
        const int nlo = (klo < K) ? (klo / d1) : 0;
        const int mhi = (khi < K) ? (khi % d1) : 0;
        const int nhi = (khi < K) ? (khi / d1) : 0;
        const float2 a = hi ? r1[mhi] : r1[mlo];     // v_cndmask, compile-time idx
        const float2 c = hi ? r2[nhi] : r2[nlo];
        aRe[t] = a.x * c.x - a.y * c.y;
        aIm[t] = a.x * c.y + a.y * c.x;
      }
#pragma unroll
      for (int ct = 0; ct < Nt; ++ct) {
        v2f bb;
        bb[0] = cgcol[ct * 16 * Kp + ks];
        bb[1] = cgcol[ct * 16 * Kp + ks + 1];
        accRe[ct] = __builtin_amdgcn_wmma_f32_16x16x4_f32(false, aRe, false, bb, (short)0, accRe[ct], false, false);
        accIm[ct] = __builtin_amdgcn_wmma_f32_16x16x4_f32(false, aIm, false, bb, (short)0, accIm[ct], false, false);
      }
    }

    // C/D layout: VGPR v -> row v (lanes 0-15) / v+8 (lanes 16-31); col = lane&15
#pragma unroll
    for (int ct = 0; ct < Nt; ++ct) {
      if (cb[ct] < 0) continue;
      const int base = cb[ct] + b * bstr[ct] + (rt * 16 + (hi << 3)) * ijstr[ct];
#pragma unroll
      for (int v = 0; v < 8; ++v) {
        v2f o; o[0] = accRe[ct][v]; o[1] = accIm[ct][v];
        __builtin_nontemporal_store(o, (v2f*)(out + base + v * ijstr[ct]));
      }
    }
  }
}

template <int P>
static inline void launch_pair(const float2* x0, const float2* x1, const float2* x2,
                               const float2* x3, const float2* x4, const float* cgw,
                               float2* out, hipStream_t stream) {
  cg_pair_kernel<P, cgt::HT.pairL1[P], cgt::HT.pairL2[P],
                 cgt::HT.pairN[P], cgt::HT.pairCgw[P]>
      <<<cgt::NB, 256, 0, stream>>>(x0, x1, x2, x3, x4, cgw, out);
}

extern "C" void kernel_launch(void* const* d_in, const int* in_sizes, int n_in,
                              void* d_out, int out_size, void* d_ws, size_t ws_size,
                              hipStream_t stream) {
  (void)in_sizes; (void)n_in; (void)out_size; (void)ws_size;
  const float2* x0 = (const float2*)d_in[0];
  const float2* x1 = (const float2*)d_in[1];
  const float2* x2 = (const float2*)d_in[2];
  const float2* x3 = (const float2*)d_in[3];
  const float2* x4 = (const float2*)d_in[4];
  const float* cg_flat = (const float*)d_in[5];
  float*  cgw = (float*)d_ws;
  float2* out = (float2*)d_out;

  const int total = cgt::HT.cgwTotal;   // 12224 floats of padded CG workspace
  cgw_zero_kernel<<<(total + 255) / 256, 256, 0, stream>>>(cgw, total);
  cgw_prep_kernel<<<cgt::NCOMBO, 256, 0, stream>>>(cg_flat, cgw);

  launch_pair<0>(x0, x1, x2, x3, x4, cgw, out, stream);
  launch_pair<1>(x0, x1, x2, x3, x4, cgw, out, stream);
  launch_pair<2>(x0, x1, x2, x3, x4, cgw, out, stream);
  launch_pair<3>(x0, x1, x2, x3, x4, cgw, out, stream);
  launch_pair<4>(x0, x1, x2, x3, x4, cgw, out, stream);
  launch_pair<5>(x0, x1, x2, x3, x4, cgw, out, stream);
  launch_pair<6>(x0, x1, x2, x3, x4, cgw, out, stream);
  launch_pair<7>(x0, x1, x2, x3, x4, cgw, out, stream);
  launch_pair<8>(x0, x1, x2, x3, x4, cgw, out, stream);
  launch_pair<9>(x0, x1, x2, x3, x4, cgw, out, stream);
  launch_pair<10>(x0, x1, x2, x3, x4, cgw, out, stream);
  launch_pair<11>(x0, x1, x2, x3, x4, cgw, out, stream);
  launch_pair<12>(x0, x1, x2, x3, x4, cgw, out, stream);
  launch_pair<13>(x0, x1, x2, x3, x4, cgw, out, stream);
  launch_pair<14>(x0, x1, x2, x3, x4, cgw, out, stream);
}